// CAM01_42167988912797
// MI455X (gfx1250) — compile-verified
//
#include <hip/hip_runtime.h>
#include <hip/hip_bf16.h>

typedef __attribute__((ext_vector_type(2))) float v2f;
typedef __attribute__((ext_vector_type(8))) float v8f;

#define HW        16384          // 128*128 pixels per batch image
#define CIN       512
#define CH        30             // conv0 output channels
#define CHP       32             // padded to 32
#define NCLS      17
#define RES_C     (NCLS * CH)    // 510
#define RES_TOTAL ((size_t)8 * RES_C * HW)
#define BN_EPS    1e-5f

// 256 threads = 8 wave32. Each wave: 32 pixels x 32(=30) channels GEMM tile,
// then per-lane (1 pixel/lane) conv1 + argmax + scatter of res/out.
__launch_bounds__(256, 2)
__global__ void CAM01_fused_kernel(const float* __restrict__ x,
                                   const float* __restrict__ w0,
                                   const float* __restrict__ b0,
                                   const float* __restrict__ gamma,
                                   const float* __restrict__ beta,
                                   const float* __restrict__ mean,
                                   const float* __restrict__ var,
                                   const float* __restrict__ w1,
                                   const float* __restrict__ b1,
                                   float* __restrict__ out)
{
    __shared__ float w0s[CIN * CHP];      // w0 transposed: [k][ch], 64 KB
    __shared__ float hTile[8][32][34];    // per-wave h staging, pad 34 -> conflict-free
    __shared__ float w1s[NCLS * CH];
    __shared__ float b1s[NCLS];
    __shared__ float bnS[CHP], bnB[CHP];  // folded BN scale / bias (conv bias folded in)

    const int tid = threadIdx.x;

    // ---- stage weights / constants into LDS ----
    for (int idx = tid; idx < CIN * CHP; idx += 256) {
        const int c = idx >> 5;           // input channel (k)
        const int o = idx & 31;           // output channel
        w0s[idx] = (o < CH) ? w0[o * CIN + c] : 0.f;
    }
    if (tid < NCLS * CH) w1s[tid] = w1[tid];
    if (tid < NCLS)      b1s[tid] = b1[tid];
    if (tid < CHP) {
        float s = 0.f, bb = 0.f;
        if (tid < CH) {
            s  = gamma[tid] * rsqrtf(var[tid] + BN_EPS);
            bb = (b0[tid] - mean[tid]) * s + beta[tid];
        }
        bnS[tid] = s; bnB[tid] = bb;
    }
    __syncthreads();

    const int wave = tid >> 5;
    const int ln   = tid & 31;
    const int ln15 = ln & 15;
    const int hi   = ln >> 4;             // lane half: 0 or 1
    const int kOff = hi << 1;             // K offset within a 16x4 tile (ISA A/B layout)

    const int b    = blockIdx.x >> 6;                      // batch
    const int pix0 = ((blockIdx.x & 63) << 8) + (wave << 5); // wave's pixel base in HW
    const float* xb = x + (size_t)b * CIN * HW;

    // A-matrix pixel pointers for the two 16-pixel M-tiles
    const float* xA0 = xb + (size_t)kOff * HW + pix0 + ln15;
    const float* xA1 = xA0 + 16;

    v8f acc00 = {}, acc01 = {}, acc10 = {}, acc11 = {};

    // ---- conv0 GEMM: K=512 in steps of 4, fp32 WMMA ----
    for (int k0 = 0; k0 < CIN; k0 += 4) {
        const float* pA0 = xA0 + (size_t)k0 * HW;
        const float* pA1 = xA1 + (size_t)k0 * HW;
        v2f a0, a1, bt0, bt1;
        a0.x = pA0[0];  a0.y = pA0[HW];     // K=k0+kOff, k0+kOff+1
        a1.x = pA1[0];  a1.y = pA1[HW];
        const int bi = (k0 + kOff) * CHP + ln15;
        bt0.x = w0s[bi];      bt0.y = w0s[bi + CHP];        // channels 0..15
        bt1.x = w0s[bi + 16]; bt1.y = w0s[bi + CHP + 16];   // channels 16..31

        acc00 = __builtin_amdgcn_wmma_f32_16x16x4_f32(false, a0, false, bt0, (short)0, acc00, false, false);
        acc01 = __builtin_amdgcn_wmma_f32_16x16x4_f32(false, a0, false, bt1, (short)0, acc01, false, false);
        acc10 = __builtin_amdgcn_wmma_f32_16x16x4_f32(false, a1, false, bt0, (short)0, acc10, false, false);
        acc11 = __builtin_amdgcn_wmma_f32_16x16x4_f32(false, a1, false, bt1, (short)0, acc11, false, false);
    }

    // ---- BN + ReLU, scatter D tiles to per-wave LDS h-tile ----
    const float s0 = bnS[ln15],      bb0 = bnB[ln15];
    const float s1 = bnS[ln15 + 16], bb1 = bnB[ln15 + 16];
    const int prow = hi << 3;   // lanes 16-31 hold rows r+8
#pragma unroll
    for (int r = 0; r < 8; ++r) {
        hTile[wave][prow + r     ][ln15     ] = fmaxf(acc00[r] * s0 + bb0, 0.f);
        hTile[wave][prow + r     ][ln15 + 16] = fmaxf(acc01[r] * s1 + bb1, 0.f);
        hTile[wave][prow + r + 16][ln15     ] = fmaxf(acc10[r] * s0 + bb0, 0.f);
        hTile[wave][prow + r + 16][ln15 + 16] = fmaxf(acc11[r] * s1 + bb1, 0.f);
    }
    // same-wave LDS ops are in-order; compiler inserts s_wait_dscnt before reads

    // ---- per-lane: one pixel -> conv1, argmax, write logits + res ----
    float h[CH];
#pragma unroll
    for (int j = 0; j < CH; ++j) h[j] = hTile[wave][ln][j];

    const int pix = pix0 + ln;  // within batch HW; consecutive across lanes -> coalesced
    float* resp = out + (size_t)b * RES_C * HW + pix;
    float* logp = out + RES_TOTAL + (size_t)b * NCLS * HW + pix;

    int best = 0; float bestv = -__builtin_inff();
#pragma unroll 1
    for (int o = 0; o < NCLS; ++o) {
        float s = b1s[o];
#pragma unroll
        for (int j = 0; j < CH; ++j) s = fmaf(w1s[o * CH + j], h[j], s);
        logp[(size_t)o * HW] = s;
        if (s > bestv) { bestv = s; best = o; }   // strict > keeps first max (jnp.argmax)
    }

#pragma unroll 1
    for (int k = 0; k < NCLS; ++k) {
        const bool sel = (k == best);
#pragma unroll
        for (int j = 0; j < CH; ++j) {
            resp[(size_t)(k * CH + j) * HW] = sel ? h[j] : 0.f;
        }
    }
}

extern "C" void kernel_launch(void* const* d_in, const int* in_sizes, int n_in,
                              void* d_out, int out_size, void* d_ws, size_t ws_size,
                              hipStream_t stream) {
    const float* x     = (const float*)d_in[0];
    const float* w0    = (const float*)d_in[1];
    const float* b0    = (const float*)d_in[2];
    const float* gamma = (const float*)d_in[3];
    const float* beta  = (const float*)d_in[4];
    const float* mean  = (const float*)d_in[5];
    const float* var   = (const float*)d_in[6];
    const float* w1    = (const float*)d_in[7];
    const float* b1    = (const float*)d_in[8];

    // 131072 pixels total / 256 per block = 512 blocks (blockIdx>>6 = batch)
    CAM01_fused_kernel<<<512, 256, 0, stream>>>(x, w0, b0, gamma, beta, mean, var,
                                                w1, b1, (float*)d_out);
}